// IRMC_GC_Model_50302656971287
// MI455X (gfx1250) — compile-verified
//
#include <hip/hip_runtime.h>
#include <hip/hip_bf16.h>
#include <math.h>

#define N_USER 100000
#define N_ITEM 50000
#define DIM    64
#define NH     4
#define NS     500
#define NB     1024
#define NL     50
#define NE     1500000
#define HID    64

typedef float v2f __attribute__((ext_vector_type(2)));
typedef float v8f __attribute__((ext_vector_type(8)));

// ======================= generic f32 WMMA GEMM =======================
// One wave (32 threads) computes one 16x16 tile of C[z] = epi(A[z]*B[z]+bias).
// A: MxK row-major (lda). B addressed as element[k,n] = Bm[k*ldbK + n*ldbN]
// (so a transpose is just a stride swap). K stepped by 4 with
// v_wmma_f32_16x16x4_f32.
// Register layouts per CDNA5 ISA 7.12.2:
//   A 16x4: lanes 0-15 -> m=lane, k={k0,k0+1}; lanes 16-31 -> m=lane-16, k={k0+2,k0+3}
//   B  4x16 (mirror): lanes 0-15 -> n=lane, k={k0,k0+1}; lanes 16-31 -> k={k0+2,k0+3}
//   C/D: vgpr v -> row m16 + (lane>>4)*8 + v, col n16 + (lane&15)
template <int EPI> // 0=none, 1=bias+relu, 2=bias+tanh
__global__ __launch_bounds__(32) void wmma_gemm_f32(
    const float* __restrict__ A, int lda, long sAz,
    const float* __restrict__ Bm, int ldbK, int ldbN, long sBz,
    const float* __restrict__ bias,
    float* __restrict__ C, int ldc, long sCz, int K)
{
  const int lane = threadIdx.x;
  const int z = blockIdx.z;
  A  += (long)z * sAz;
  Bm += (long)z * sBz;
  C  += (long)z * sCz;
  const int m16 = blockIdx.x * 16, n16 = blockIdx.y * 16;
  const int half = lane >> 4;          // 0 or 1
  const int mA = m16 + (lane & 15);
  const int nB = n16 + (lane & 15);
  v8f acc = {};
  for (int k0 = 0; k0 < K; k0 += 4) {
    const int k = k0 + half * 2;
    v2f a, b;
    a[0] = A[(long)mA * lda + k];
    a[1] = A[(long)mA * lda + k + 1];
    b[0] = Bm[(long)k * ldbK + (long)nB * ldbN];
    b[1] = Bm[(long)(k + 1) * ldbK + (long)nB * ldbN];
    acc = __builtin_amdgcn_wmma_f32_16x16x4_f32(false, a, false, b,
                                                (short)0, acc, false, false);
  }
  const int ccol  = n16 + (lane & 15);
  const int crow0 = m16 + half * 8;
  float bv = 0.0f;
  if (EPI != 0) bv = bias[ccol];
#pragma unroll
  for (int v = 0; v < 8; ++v) {
    float val = acc[v];
    if (EPI == 1) val = fmaxf(val + bv, 0.0f);
    if (EPI == 2) val = tanhf(val + bv);
    C[(long)(crow0 + v) * ldc + ccol] = val;
  }
}

// ======================= helper kernels =======================

__global__ void init_slots_kernel(int* user_slot, int* item_slot) {
  int i = blockIdx.x * blockDim.x + threadIdx.x;
  if (i < N_USER) user_slot[i] = 0x7FFFFFFF;
  if (i < N_ITEM) item_slot[i] = 0x7FFFFFFF;
}

__global__ void zero_f_kernel(float* p, int n) {
  int i = blockIdx.x * blockDim.x + threadIdx.x;
  if (i < n) p[i] = 0.0f;
}

// Canonical slot = min batch index among rows carrying this user/item
// (deterministic under duplicates).
__global__ void set_slots_kernel(const int* __restrict__ x,
                                 int* user_slot, int* item_slot) {
  int b = blockIdx.x * blockDim.x + threadIdx.x;
  if (b < NB) {
    atomicMin(&user_slot[x[2 * b]], b);
    atomicMin(&item_slot[x[2 * b + 1]], b);
  }
}

// Filtered segment-sum: only ~2% of edges touch a batch user/item.
__global__ void edge_scatter_kernel(const int* __restrict__ eu,
                                    const int* __restrict__ ei,
                                    const int* __restrict__ user_slot,
                                    const int* __restrict__ item_slot,
                                    const float* __restrict__ user_emb_tab,
                                    const float* __restrict__ item_emb_tab,
                                    float* agg_item, float* agg_user,
                                    float* deg_u, float* deg_i) {
  int e = blockIdx.x * blockDim.x + threadIdx.x;
  if (e >= NE) return;
  int u = eu[e], it = ei[e];
  int su = user_slot[u];
  if (su < NB) {
    atomicAdd(&deg_u[su], 1.0f);
    const float* src = item_emb_tab + (long)it * DIM;
    float* dst = agg_item + (long)su * DIM;
    for (int d = 0; d < DIM; ++d) atomicAdd(&dst[d], src[d]);
  }
  int si = item_slot[it];
  if (si < NB) {
    atomicAdd(&deg_i[si], 1.0f);
    const float* src = user_emb_tab + (long)u * DIM;
    float* dst = agg_user + (long)si * DIM;
    for (int d = 0; d < DIM; ++d) atomicAdd(&dst[d], src[d]);
  }
}

// user_init_emb = sum over all L history items / history_len
__global__ void uinit_kernel(const int* __restrict__ history,
                             const int* __restrict__ history_len,
                             const float* __restrict__ item_emb_tab,
                             float* __restrict__ uinit) {
  __shared__ int h[NL];
  int b = blockIdx.x, t = threadIdx.x;
  if (t < NL) h[t] = history[b * NL + t];
  __syncthreads();
  float acc = 0.0f;
  for (int l = 0; l < NL; ++l) acc += item_emb_tab[(long)h[l] * DIM + t];
  uinit[(long)b * DIM + t] = acc / (float)history_len[b];
}

// logits[h,b,s] = samp_emb[h,b,s,:] . qk[h,b,:]   (qk = uinit @ (wq wk^T))
__global__ void attn_logits_kernel(const int* __restrict__ sample_index,
                                   const int* __restrict__ supp_users,
                                   const float* __restrict__ user_emb_tab,
                                   const float* __restrict__ qk,
                                   float* __restrict__ logits) {
  __shared__ float qv[DIM];
  int hb = blockIdx.x;           // h*NB + b
  if (threadIdx.x < DIM) qv[threadIdx.x] = qk[(long)hb * DIM + threadIdx.x];
  __syncthreads();
  int s = threadIdx.x;
  if (s < NS) {
    int idx = supp_users[sample_index[(long)hb * NS + s]];
    const float4* r4 = (const float4*)(user_emb_tab + (long)idx * DIM);
    float acc = 0.0f;
#pragma unroll
    for (int j = 0; j < DIM / 4; ++j) {
      float4 v = r4[j];
      acc += v.x * qv[4 * j] + v.y * qv[4 * j + 1] +
             v.z * qv[4 * j + 2] + v.w * qv[4 * j + 3];
    }
    logits[(long)hb * NS + s] = acc;
  }
}

// softmax over S then ctx_pre[h,b,d] = sum_s attn_s * samp_emb[h,b,s,d]
__global__ void attn_softmax_ctx_kernel(const int* __restrict__ sample_index,
                                        const int* __restrict__ supp_users,
                                        const float* __restrict__ user_emb_tab,
                                        const float* __restrict__ logits,
                                        float* __restrict__ ctx_pre) {
  __shared__ float wsh[512];
  __shared__ int   sidx[512];
  __shared__ float red[64];
  int hb = blockIdx.x, t = threadIdx.x;   // blockDim = 64
  float mx = -1e30f;
  for (int s = t; s < NS; s += 64) {
    float v = logits[(long)hb * NS + s];
    wsh[s] = v;
    mx = fmaxf(mx, v);
    sidx[s] = supp_users[sample_index[(long)hb * NS + s]];
  }
  red[t] = mx;
  for (int off = 32; off; off >>= 1) {
    __syncthreads();
    if (t < off) red[t] = fmaxf(red[t], red[t + off]);
  }
  __syncthreads();
  mx = red[0];
  __syncthreads();
  float sum = 0.0f;
  for (int s = t; s < NS; s += 64) {
    float e = __expf(wsh[s] - mx);
    wsh[s] = e;
    sum += e;
  }
  red[t] = sum;
  for (int off = 32; off; off >>= 1) {
    __syncthreads();
    if (t < off) red[t] += red[t + off];
  }
  __syncthreads();
  float inv = 1.0f / red[0];
  // thread t = dimension d; rows are L2-resident (SUPP*256B = 2.6MB)
  float acc = 0.0f;
  for (int s = 0; s < NS; ++s)
    acc += wsh[s] * user_emb_tab[(long)sidx[s] * DIM + t];
  ctx_pre[(long)hb * DIM + t] = acc * inv;
}

// gcn_*_h = agg/deg+1 gathered at each batch row via the slot maps
__global__ void normalize_h_kernel(const int* __restrict__ x,
                                   const int* __restrict__ user_slot,
                                   const int* __restrict__ item_slot,
                                   const float* __restrict__ agg_item,
                                   const float* __restrict__ agg_user,
                                   const float* __restrict__ deg_u,
                                   const float* __restrict__ deg_i,
                                   float* __restrict__ gcn_item_h,
                                   float* __restrict__ gcn_user_h) {
  int b = blockIdx.x, d = threadIdx.x;
  int su = user_slot[x[2 * b]];
  int si = item_slot[x[2 * b + 1]];
  gcn_item_h[(long)b * DIM + d] = agg_item[(long)su * DIM + d] / (deg_u[su] + 1.0f);
  gcn_user_h[(long)b * DIM + d] = agg_user[(long)si * DIM + d] / (deg_i[si] + 1.0f);
}

__global__ void feats_kernel(const int* __restrict__ x,
                             const float* __restrict__ user_emb,
                             const float* __restrict__ item_emb_tab,
                             const float* __restrict__ gcn_user_out,
                             const float* __restrict__ gcn_item_out,
                             float* __restrict__ feats) {
  int b = blockIdx.x, j = threadIdx.x;   // blockDim = 256
  int d = j & 63, sec = j >> 6;
  float ue = user_emb[(long)b * DIM + d];
  float ie = item_emb_tab[(long)x[2 * b + 1] * DIM + d];
  float gu = gcn_user_out[(long)b * DIM + d];
  float gi = gcn_item_out[(long)b * DIM + d];
  float v = (sec == 0) ? ue * ie : (sec == 1) ? ue * gi
          : (sec == 2) ? gu * ie : gu * gi;
  feats[(long)b * 4 * DIM + j] = v;
}

__global__ void final_kernel(const int* __restrict__ x,
                             const float* __restrict__ h2,
                             const float* __restrict__ l3_w,
                             const float* __restrict__ l3_b,
                             const float* __restrict__ user_bias,
                             const float* __restrict__ item_bias,
                             float* __restrict__ out) {
  int b = blockIdx.x * blockDim.x + threadIdx.x;
  if (b < NB) {
    float acc = l3_b[0];
    for (int d = 0; d < HID; ++d) acc += h2[(long)b * HID + d] * l3_w[d];
    out[b] = acc + user_bias[x[2 * b]] + item_bias[x[2 * b + 1]];
  }
}

// ======================= launcher =======================
extern "C" void kernel_launch(void* const* d_in, const int* in_sizes, int n_in,
                              void* d_out, int out_size, void* d_ws, size_t ws_size,
                              hipStream_t stream) {
  const int*   x            = (const int*)d_in[0];
  const int*   history      = (const int*)d_in[1];
  const int*   history_len  = (const int*)d_in[2];
  const int*   sample_index = (const int*)d_in[3];
  const int*   supp_users   = (const int*)d_in[4];
  const int*   edge_index   = (const int*)d_in[5];
  const float* user_emb_tab = (const float*)d_in[6];
  const float* item_emb_tab = (const float*)d_in[7];
  const float* wq           = (const float*)d_in[8];
  const float* wk           = (const float*)d_in[9];
  const float* wv           = (const float*)d_in[10];
  const float* w_out_w      = (const float*)d_in[11];
  const float* gcn_user_w   = (const float*)d_in[12];
  const float* gcn_user_b   = (const float*)d_in[13];
  const float* gcn_item_w   = (const float*)d_in[14];
  const float* gcn_item_b   = (const float*)d_in[15];
  const float* l1_w         = (const float*)d_in[16];
  const float* l1_b         = (const float*)d_in[17];
  const float* l2_w         = (const float*)d_in[18];
  const float* l2_b         = (const float*)d_in[19];
  const float* l3_w         = (const float*)d_in[20];
  const float* l3_b         = (const float*)d_in[21];
  const float* user_bias    = (const float*)d_in[22];
  const float* item_bias    = (const float*)d_in[23];
  float* out = (float*)d_out;

  // ---- workspace carve-out (~16.5 MB) ----
  char* base = (char*)d_ws;
  size_t off = 0;
  auto alloc = [&](size_t bytes) -> char* {
    char* p = base + off;
    off += (bytes + 255) & ~(size_t)255;
    return p;
  };
  int*   user_slot  = (int*)alloc(N_USER * sizeof(int));
  int*   item_slot  = (int*)alloc(N_ITEM * sizeof(int));
  float* agg_item   = (float*)alloc((2 * NB * DIM + 2 * NB) * sizeof(float));
  float* agg_user   = agg_item + NB * DIM;
  float* deg_u      = agg_item + 2 * NB * DIM;
  float* deg_i      = deg_u + NB;
  float* uinit      = (float*)alloc(NB * DIM * sizeof(float));
  float* wqk        = (float*)alloc(NH * DIM * DIM * sizeof(float));   // wq @ wk^T
  float* wkv        = (float*)alloc(NH * DIM * DIM * sizeof(float));   // wk @ wv
  float* qk         = (float*)alloc((size_t)NH * NB * DIM * sizeof(float));
  float* logits     = (float*)alloc((size_t)NH * NB * NS * sizeof(float));
  float* ctx_pre    = (float*)alloc((size_t)NH * NB * DIM * sizeof(float));
  float* gat        = (float*)alloc((size_t)NB * NH * DIM * sizeof(float)); // [b][h*64+f]
  float* user_emb   = (float*)alloc(NB * DIM * sizeof(float));
  float* gcn_item_h = (float*)alloc(NB * DIM * sizeof(float));
  float* gcn_user_h = (float*)alloc(NB * DIM * sizeof(float));
  float* gcn_user_o = (float*)alloc(NB * DIM * sizeof(float));
  float* gcn_item_o = (float*)alloc(NB * DIM * sizeof(float));
  float* feats      = (float*)alloc((size_t)NB * 4 * DIM * sizeof(float));
  float* h1         = (float*)alloc((size_t)NB * 2 * HID * sizeof(float));
  float* h2         = (float*)alloc((size_t)NB * HID * sizeof(float));
  (void)ws_size; (void)n_in; (void)in_sizes; (void)out_size;

  const int* eu = edge_index;       // row 0
  const int* ei = edge_index + NE;  // row 1

  // 1) slot maps + zero accumulators
  init_slots_kernel<<<(N_USER + 255) / 256, 256, 0, stream>>>(user_slot, item_slot);
  int nz = 2 * NB * DIM + 2 * NB;
  zero_f_kernel<<<(nz + 255) / 256, 256, 0, stream>>>(agg_item, nz);
  set_slots_kernel<<<(NB + 255) / 256, 256, 0, stream>>>(x, user_slot, item_slot);

  // 2) filtered GCN segment-sum over edges
  edge_scatter_kernel<<<(NE + 255) / 256, 256, 0, stream>>>(
      eu, ei, user_slot, item_slot, user_emb_tab, item_emb_tab,
      agg_item, agg_user, deg_u, deg_i);

  // 3) user_init_emb
  uinit_kernel<<<NB, DIM, 0, stream>>>(history, history_len, item_emb_tab, uinit);

  // 4) small weight products: wqk[h] = wq[h] @ wk[h]^T, wkv[h] = wk[h] @ wv[h]
  wmma_gemm_f32<0><<<dim3(4, 4, NH), 32, 0, stream>>>(
      wq, DIM, DIM * DIM, wk, /*ldbK=*/1, /*ldbN=*/DIM, DIM * DIM,
      nullptr, wqk, DIM, DIM * DIM, DIM);
  wmma_gemm_f32<0><<<dim3(4, 4, NH), 32, 0, stream>>>(
      wk, DIM, DIM * DIM, wv, /*ldbK=*/DIM, /*ldbN=*/1, DIM * DIM,
      nullptr, wkv, DIM, DIM * DIM, DIM);

  // 5) qk[h] = uinit @ wqk[h]   (A shared across heads: sAz = 0)
  wmma_gemm_f32<0><<<dim3(NB / 16, DIM / 16, NH), 32, 0, stream>>>(
      uinit, DIM, 0, wqk, DIM, 1, DIM * DIM,
      nullptr, qk, DIM, (long)NB * DIM, DIM);

  // 6) attention: logits, softmax, ctx_pre = attn @ samp_emb
  attn_logits_kernel<<<NH * NB, 512, 0, stream>>>(
      sample_index, supp_users, user_emb_tab, qk, logits);
  attn_softmax_ctx_kernel<<<NH * NB, 64, 0, stream>>>(
      sample_index, supp_users, user_emb_tab, logits, ctx_pre);

  // 7) gat[b, h*64+f] = ctx_pre[h] @ wkv[h]  (sCz=64 walks the column blocks)
  wmma_gemm_f32<0><<<dim3(NB / 16, DIM / 16, NH), 32, 0, stream>>>(
      ctx_pre, DIM, (long)NB * DIM, wkv, DIM, 1, DIM * DIM,
      nullptr, gat, NH * DIM, DIM, DIM);

  // 8) user_emb = gat @ w_out_w   (K = 256)
  wmma_gemm_f32<0><<<dim3(NB / 16, DIM / 16, 1), 32, 0, stream>>>(
      gat, NH * DIM, 0, w_out_w, DIM, 1, 0,
      nullptr, user_emb, DIM, 0, NH * DIM);

  // 9) GCN normalize + projections (bias + relu)
  normalize_h_kernel<<<NB, DIM, 0, stream>>>(
      x, user_slot, item_slot, agg_item, agg_user, deg_u, deg_i,
      gcn_item_h, gcn_user_h);
  wmma_gemm_f32<1><<<dim3(NB / 16, DIM / 16, 1), 32, 0, stream>>>(
      gcn_user_h, DIM, 0, gcn_user_w, DIM, 1, 0,
      gcn_user_b, gcn_user_o, DIM, 0, DIM);
  wmma_gemm_f32<1><<<dim3(NB / 16, DIM / 16, 1), 32, 0, stream>>>(
      gcn_item_h, DIM, 0, gcn_item_w, DIM, 1, 0,
      gcn_item_b, gcn_item_o, DIM, 0, DIM);

  // 10) feature cross + MLP (tanh epilogues)
  feats_kernel<<<NB, 4 * DIM, 0, stream>>>(
      x, user_emb, item_emb_tab, gcn_user_o, gcn_item_o, feats);
  wmma_gemm_f32<2><<<dim3(NB / 16, (2 * HID) / 16, 1), 32, 0, stream>>>(
      feats, 4 * DIM, 0, l1_w, 2 * HID, 1, 0,
      l1_b, h1, 2 * HID, 0, 4 * DIM);
  wmma_gemm_f32<2><<<dim3(NB / 16, HID / 16, 1), 32, 0, stream>>>(
      h1, 2 * HID, 0, l2_w, HID, 1, 0,
      l2_b, h2, HID, 0, 2 * HID);

  // 11) final dot + biases
  final_kernel<<<(NB + 255) / 256, 256, 0, stream>>>(
      x, h2, l3_w, l3_b, user_bias, item_bias, out);
}